// QueryAndGroupPyramid_85323820302741
// MI455X (gfx1250) — compile-verified
//
#include <hip/hip_runtime.h>
#include <hip/hip_bf16.h>

typedef float v2f __attribute__((ext_vector_type(2)));
typedef float v8f __attribute__((ext_vector_type(8)));

// ---------------------------------------------------------------------------
// Prep: pack points as (-2x,-2y,-2z,|p|^2) and queries as (x,y,z,|q|^2-r^2)
// so a single V_WMMA_F32_16X16X4_F32 produces d^2 - r^2 per (query,point).
// ---------------------------------------------------------------------------
__global__ void qgp_prep_kernel(const float* __restrict__ xyz, int Ntot,
                                const float* __restrict__ new_xyz,
                                const float* __restrict__ new_r, int Mtot,
                                float4* __restrict__ ppack,
                                float4* __restrict__ qprep) {
    int i = blockIdx.x * blockDim.x + threadIdx.x;
    if (i < Ntot) {
        float x = xyz[3 * i + 0], y = xyz[3 * i + 1], z = xyz[3 * i + 2];
        ppack[i] = make_float4(-2.0f * x, -2.0f * y, -2.0f * z, x * x + y * y + z * z);
    }
    if (i < Mtot) {
        float x = new_xyz[3 * i + 0], y = new_xyz[3 * i + 1], z = new_xyz[3 * i + 2];
        float r = new_r[i];
        qprep[i] = make_float4(x, y, z, (x * x + y * y + z * z) - r * r);
    }
}

// ---------------------------------------------------------------------------
// Segmented ball query: one wave32 = (16-query tile) x (point segment).
// WMMA f32 16x16x4 -> 16x16 tile of (d^2 - r^2); within <=> result < 0.
// 8 ballots -> uniform SGPR words; SALU OR-reduce skips all-miss tiles
// (the common case); otherwise one cndmask chain (ball[lane&7]) extracts
// each query's 16-bit mask. Software-pipelined point-tile loads (next tile's
// b128 load issued before processing the current one) hide L2 latency.
// ---------------------------------------------------------------------------
__global__ void qgp_ball_seg_kernel(const float4* __restrict__ qprep,
                                    const float4* __restrict__ ppack,
                                    const int* __restrict__ xyz_bcnt,
                                    const int* __restrict__ nq_bcnt,
                                    int B, int Mtot, int ns, int nqt, int nseg,
                                    int* __restrict__ seg_idx,
                                    int* __restrict__ seg_cnt) {
    const int lane = threadIdx.x & 31;
    const int wid  = (blockIdx.x * blockDim.x + threadIdx.x) >> 5;
    const int qt   = wid % nqt;
    const int seg  = wid / nqt;
    if (seg >= nseg) return;
    const int q0 = qt * 16;
    if (q0 >= Mtot) return;

    // batch of this query tile (tiles assumed not to straddle batch bounds)
    int b = 0, acc = 0;
    for (int bb = 0; bb < B; ++bb) { int c = nq_bcnt[bb]; if (q0 < acc + c) { b = bb; break; } acc += c; }
    int ps = 0;
    for (int bb = 0; bb < b; ++bb) ps += xyz_bcnt[bb];
    const int Npts = xyz_bcnt[b];

    const int  m      = lane & 15;
    const bool lo     = lane < 16;
    const int  q      = q0 + lane;               // query owned by lanes 0..15
    const bool active = lo && (q < Mtot);

    // 16-aligned segment [s0, s1) of this batch's point range
    const int segLen = ((((Npts + nseg - 1) / nseg) + 15) >> 4) << 4;
    int s0 = seg * segLen; if (s0 > Npts) s0 = Npts;
    int s1 = s0 + segLen;  if (s1 > Npts) s1 = Npts;

    int cnt = 0;
    if (s1 > s0) {
        __builtin_prefetch(&ppack[ps + s0 + m], 0, 0);   // warm the stream

        // A-matrix (16x4 f32): lanes 0-15 -> K=0,1 ; lanes 16-31 -> K=2,3
        int qa = q0 + m; if (qa >= Mtot) qa = Mtot - 1;
        const float4 myq = qprep[qa];
        v2f a;
        a[0] = lo ? myq.x : myq.z;
        a[1] = lo ? myq.y : 1.0f;

        // C-matrix: VGPR r holds M=r (lanes 0-15) and M=r+8 (lanes 16-31)
        v8f cvec;
#pragma unroll
        for (int r = 0; r < 8; ++r) {
            int qm = q0 + (lo ? r : r + 8);
            if (qm >= Mtot) qm = Mtot - 1;
            cvec[r] = qprep[qm].w;               // |q|^2 - r^2
        }

        int* op = seg_idx + ((size_t)seg * Mtot + q) * ns;
        const int nsub = (s1 - s0 + 15) >> 4;    // subtiles incl. partial tail
        const int rem  = (s1 - s0) & 15;
        const int pmax = s1 - 1;                 // clamp for last subtile

        // software-pipelined: pp holds tile t's data, next load in flight
        int pidx = s0 + m; if (pidx > pmax) pidx = pmax;
        float4 pp = ppack[ps + pidx];

        for (int t = 0; t < nsub; ++t) {
            const int base = s0 + (t << 4);
            const float4 cur = pp;
            // issue next tile's load early (clamped -> branch-free)
            int nidx = base + 16 + m; if (nidx > pmax) nidx = pmax;
            pp = ppack[ps + nidx];

            // B-matrix (4x16 f32): lanes 0-15 -> K=0,1 ; lanes 16-31 -> K=2,3
            v2f bm;
            bm[0] = lo ? cur.x : cur.z;
            bm[1] = lo ? cur.y : cur.w;

            // D[q][p] = |p|^2 - 2 q.p + |q|^2 - r^2 = d^2 - r^2
            v8f d = __builtin_amdgcn_wmma_f32_16x16x4_f32(
                false, a, false, bm, (short)0, cvec, false, false);

            unsigned ball[8];
#pragma unroll
            for (int r = 0; r < 8; ++r)
                ball[r] = (unsigned)__ballot(d[r] < 0.0f);

            const unsigned anyb = ball[0] | ball[1] | ball[2] | ball[3] |
                                  ball[4] | ball[5] | ball[6] | ball[7];
            if (anyb) {                          // uniform branch, rare
                unsigned sel  = ball[lane & 7];  // single select chain
                unsigned mask = (lane & 8) ? (sel >> 16) : (sel & 0xffffu);
                if (t == nsub - 1 && rem > 0)    // partial tail subtile
                    mask &= (1u << rem) - 1u;
                if (active) {
                    while (mask && cnt < ns) {
                        int bit = __builtin_ctz(mask);
                        op[cnt++] = base + bit;  // batch-local point index
                        mask &= mask - 1;
                    }
                }
                if (__ballot(active && cnt < ns) == 0ull) break;
            }
        }
    }
    if (active) seg_cnt[(size_t)seg * Mtot + q] = cnt;
}

// ---------------------------------------------------------------------------
// Merge: ordered concat of per-segment first-ns lists -> exact global
// first-ns by index; pad with first index; empty flag.
// ---------------------------------------------------------------------------
__global__ void qgp_merge_kernel(const int* __restrict__ seg_idx,
                                 const int* __restrict__ seg_cnt,
                                 int nseg, int Mtot, int ns,
                                 int* __restrict__ idx_out,
                                 int* __restrict__ empty_out) {
    const int q = blockIdx.x * blockDim.x + threadIdx.x;
    if (q >= Mtot) return;
    int total = 0, first = 0;
    for (int s = 0; s < nseg && total < ns; ++s) {
        const int c = seg_cnt[(size_t)s * Mtot + q];
        const int* sp = seg_idx + ((size_t)s * Mtot + q) * ns;
        for (int j = 0; j < c && total < ns; ++j) {
            int v = sp[j];
            if (total == 0) first = v;
            idx_out[(size_t)q * ns + total] = v;
            ++total;
        }
    }
    empty_out[q] = (total == 0) ? 1 : 0;
    for (int t = total; t < ns; ++t)
        idx_out[(size_t)q * ns + t] = first;
}

// ---------------------------------------------------------------------------
// Grouping: one wave per query; lane = sample slot -> 128B coalesced stores
// per output row. Feature gathers are float4 and L2-resident (8MB << 192MB).
// ---------------------------------------------------------------------------
__global__ void qgp_group_kernel(const float* __restrict__ xyz,
                                 const float* __restrict__ new_xyz,
                                 const float* __restrict__ features,
                                 const int* __restrict__ xyz_bcnt,
                                 const int* __restrict__ nq_bcnt,
                                 const int* __restrict__ idx_in,
                                 const int* __restrict__ empty_in,
                                 int B, int Mtot, int C, int ns,
                                 float* __restrict__ out) {
    const int lane = threadIdx.x & 31;
    const int q = blockIdx.x * (blockDim.x >> 5) + (threadIdx.x >> 5);
    if (q >= Mtot) return;

    int b = 0, acc = 0;
    for (int bb = 0; bb < B; ++bb) { int c = nq_bcnt[bb]; if (q < acc + c) { b = bb; break; } acc += c; }
    int ps = 0;
    for (int bb = 0; bb < b; ++bb) ps += xyz_bcnt[bb];

    const int empty = empty_in[q];
    const float qx = new_xyz[3 * q + 0];
    const float qy = new_xyz[3 * q + 1];
    const float qz = new_xyz[3 * q + 2];
    float* ob = out + (size_t)q * (size_t)(3 + C) * (size_t)ns;

    for (int s = lane; s < ns; s += 32) {
        const int il  = idx_in[(size_t)q * ns + s];
        const int pid = ps + il;

        float gx = empty ? 0.0f : (xyz[3 * pid + 0] - qx);
        float gy = empty ? 0.0f : (xyz[3 * pid + 1] - qy);
        float gz = empty ? 0.0f : (xyz[3 * pid + 2] - qz);
        ob[(size_t)0 * ns + s] = gx;
        ob[(size_t)1 * ns + s] = gy;
        ob[(size_t)2 * ns + s] = gz;

        if ((C & 3) == 0) {
            const float4* f4 = (const float4*)(features + (size_t)pid * C);
            for (int c = 0; c < C; c += 4) {
                float4 v = f4[c >> 2];
                if (empty) v = make_float4(0.f, 0.f, 0.f, 0.f);
                ob[(size_t)(3 + c + 0) * ns + s] = v.x;
                ob[(size_t)(3 + c + 1) * ns + s] = v.y;
                ob[(size_t)(3 + c + 2) * ns + s] = v.z;
                ob[(size_t)(3 + c + 3) * ns + s] = v.w;
            }
        } else {
            for (int c = 0; c < C; ++c) {
                float v = empty ? 0.0f : features[(size_t)pid * C + c];
                ob[(size_t)(3 + c) * ns + s] = v;
            }
        }
    }
}

// ---------------------------------------------------------------------------
extern "C" void kernel_launch(void* const* d_in, const int* in_sizes, int n_in,
                              void* d_out, int out_size, void* d_ws, size_t ws_size,
                              hipStream_t stream) {
    const float* xyz  = (const float*)d_in[0];
    const int*   xbc  = (const int*)d_in[1];
    const float* nxyz = (const float*)d_in[2];
    const float* nr   = (const float*)d_in[3];
    const int*   nbc  = (const int*)d_in[4];
    const float* feat = (const float*)d_in[5];

    const int Ntot = in_sizes[0] / 3;          // total points (all batches)
    const int B    = in_sizes[1];
    const int Mtot = in_sizes[3];              // total queries (all batches)
    const int C    = in_sizes[5] / Ntot;
    // out = Mtot*(3+C)*ns floats + Mtot*ns ints  ->  ns = out_size/(Mtot*(C+4))
    const int ns   = out_size / (Mtot * (C + 4));

    float* nf   = (float*)d_out;
    int*   idxo = (int*)(nf + (size_t)Mtot * (size_t)(3 + C) * (size_t)ns);

    // Workspace: ppack | qprep | emptyf | seg_cnt | seg_idx
    const size_t fixed = (size_t)Ntot * 16 + (size_t)Mtot * 16 + (size_t)Mtot * 4;
    int nseg = 4;
    while (nseg > 1 &&
           fixed + (size_t)nseg * (size_t)Mtot * (size_t)(ns + 1) * 4 > ws_size)
        nseg >>= 1;

    char*   ws      = (char*)d_ws;
    float4* ppack   = (float4*)ws;
    float4* qprep   = (float4*)(ws + (size_t)Ntot * 16);
    int*    emptyf  = (int*)(ws + (size_t)Ntot * 16 + (size_t)Mtot * 16);
    int*    segcnt  = emptyf + Mtot;
    int*    segidx  = segcnt + (size_t)nseg * Mtot;

    const int nmax = (Ntot > Mtot) ? Ntot : Mtot;
    qgp_prep_kernel<<<(nmax + 255) / 256, 256, 0, stream>>>(
        xyz, Ntot, nxyz, nr, Mtot, ppack, qprep);

    const int nqt   = (Mtot + 15) / 16;        // query tiles
    const int waves = nqt * nseg;              // one wave per (tile, segment)
    qgp_ball_seg_kernel<<<(waves + 1) / 2, 64, 0, stream>>>(
        qprep, ppack, xbc, nbc, B, Mtot, ns, nqt, nseg, segidx, segcnt);

    qgp_merge_kernel<<<(Mtot + 255) / 256, 256, 0, stream>>>(
        segidx, segcnt, nseg, Mtot, ns, idxo, emptyf);

    qgp_group_kernel<<<(Mtot + 7) / 8, 256, 0, stream>>>(
        xyz, nxyz, feat, xbc, nbc, idxo, emptyf, B, Mtot, C, ns, nf);
}